// DecoderViT_17403207483577
// MI455X (gfx1250) — compile-verified
//
#include <hip/hip_runtime.h>
#include <math.h>

// ---------------------------------------------------------------------------
// CDNA5 (gfx1250) fused SPT-ViT implementation, f32 WMMA (16x16x4) everywhere.
// All GEMMs are NT (C = A * B^T), matching the f32 WMMA operand layout:
//   A frag: lane l holds A[row=l&15][k = 4*kk + 2*(l>>4) + {0,1}]  (float2)
//   B frag: lane l holds B[row=l&15][k = 4*kk + 2*(l>>4) + {0,1}]  (float2)
//   C/D   : vgpr r, lane l -> C[row = r + 8*(l>>4)][col = l&15]
// OOB rows are CLAMPED (not predicated) -> branch-free global_load_b64 loops.
// Fragments are preloaded in batches ahead of the WMMA chains so loads form
// clauses instead of load->wait->wmma serialization.
// ---------------------------------------------------------------------------

typedef float v2f __attribute__((ext_vector_type(2)));
typedef float v8f __attribute__((ext_vector_type(8)));

#define NTOK   841          // 29*29 patches
#define HP     29
#define DIM    256
#define BATCH  8
#define MTOT   (BATCH * NTOK)   // 6728 token rows
#define NHEADS 8
#define DH     32
#define NEGV   (-1e30f)
#define IMGSZ  457

static __device__ __forceinline__ v8f wmma_f32(v2f a, v2f b, v8f c) {
  return __builtin_amdgcn_wmma_f32_16x16x4_f32(false, a, false, b, (short)0, c,
                                               false, false);
}

// ---------------------------------------------------------------------------
// 1) Shifted-patch gather: builds U[MTOT][1280] from x[8][457][457].
// ---------------------------------------------------------------------------
__global__ void build_u_kernel(const float* __restrict__ x, float* __restrict__ U) {
  int idx = blockIdx.x * 256 + threadIdx.x;
  const int total = MTOT * 1280;
  if (idx >= total) return;
  int f = idx % 1280;
  int m = idx / 1280;
  int b = m / NTOK;
  int n = m % NTOK;
  int hp = n / HP, wp = n % HP;
  int kch = f >> 8;            // which shift (0..4)
  int rc = f & 255;
  int r = rc >> 4, cc = rc & 15;
  const int dxs[5] = {0, -8, 8, -8, 8};
  const int dys[5] = {0, -8, -8, 8, 8};
  int i = hp * 16 + r - dys[kch];   // (i - dy) mod 464
  int j = wp * 16 + cc - dxs[kch];
  if (i < 0) i += 464; else if (i >= 464) i -= 464;
  if (j < 0) j += 464; else if (j >= 464) j -= 464;
  float v = 0.f;
  if (i < IMGSZ && j < IMGSZ)
    v = x[((long)b * IMGSZ + i) * IMGSZ + j];
  U[idx] = v;
}

// ---------------------------------------------------------------------------
// 2) NT WMMA GEMM: C[M][N] = A[M][K] * B[N][K]^T (+ epilogue)
//    Block = 256 thr = 8 waves (2m x 4n), wave macro-tile 32x32 (2x2 WMMA
//    tiles, A/B fragments reused) -> block tile 64x128. N % 128, K % 8 == 0.
//    EP: 0 none | 1 +bias | 2 +bias+pos_emb | 3 +bias+residual | 4 gelu(x+bias)
// ---------------------------------------------------------------------------
template <int EP>
__global__ void gemm_nt_kernel(const float* __restrict__ A,
                               const float* __restrict__ Bm,
                               float* __restrict__ C,
                               const float* __restrict__ bias,
                               const float* __restrict__ extra,
                               int M, int N, int K) {
  const int lane = threadIdx.x & 31;
  const int wave = threadIdx.x >> 5;
  const int wm = wave & 1;
  const int wn = wave >> 1;
  const int m0 = blockIdx.x * 64 + wm * 32;
  const int n0 = blockIdx.y * 128 + wn * 32;
  const int mr = lane & 15;
  const int g  = lane >> 4;

  int ar0 = m0 + mr;       if (ar0 > M - 1) ar0 = M - 1;   // clamp: safe
  int ar1 = m0 + 16 + mr;  if (ar1 > M - 1) ar1 = M - 1;
  const int br0 = n0 + mr;         // always < N (N % 128 == 0)
  const int br1 = n0 + 16 + mr;
  const float* ap0 = A  + (long)ar0 * K;
  const float* ap1 = A  + (long)ar1 * K;
  const float* bp0 = Bm + (long)br0 * K;
  const float* bp1 = Bm + (long)br1 * K;

  v8f c00 = {0.f,0.f,0.f,0.f,0.f,0.f,0.f,0.f};
  v8f c01 = c00, c10 = c00, c11 = c00;

  // 2 K-steps per iteration: 8 b64 loads (clause) then 8 WMMAs.
  for (int k = 2 * g; k < K; k += 8) {
    v2f a0 = *(const v2f*)(ap0 + k);
    v2f a1 = *(const v2f*)(ap1 + k);
    v2f b0 = *(const v2f*)(bp0 + k);
    v2f b1 = *(const v2f*)(bp1 + k);
    v2f a2 = *(const v2f*)(ap0 + k + 4);
    v2f a3 = *(const v2f*)(ap1 + k + 4);
    v2f b2 = *(const v2f*)(bp0 + k + 4);
    v2f b3 = *(const v2f*)(bp1 + k + 4);
    c00 = wmma_f32(a0, b0, c00);
    c01 = wmma_f32(a0, b1, c01);
    c10 = wmma_f32(a1, b0, c10);
    c11 = wmma_f32(a1, b1, c11);
    c00 = wmma_f32(a2, b2, c00);
    c01 = wmma_f32(a2, b3, c01);
    c10 = wmma_f32(a3, b2, c10);
    c11 = wmma_f32(a3, b3, c11);
  }

  v8f acc[2][2] = {{c00, c01}, {c10, c11}};
#pragma unroll
  for (int ti = 0; ti < 2; ++ti) {
#pragma unroll
    for (int tj = 0; tj < 2; ++tj) {
      const int nn = n0 + tj * 16 + mr;
#pragma unroll
      for (int r = 0; r < 8; ++r) {
        int mm = m0 + ti * 16 + r + 8 * g;
        if (mm < M) {
          float v = acc[ti][tj][r];
          if (EP >= 1) v += bias[nn];
          if (EP == 2) v += extra[(long)(mm % NTOK) * N + nn];   // pos_emb
          if (EP == 3) v += extra[(long)mm * N + nn];            // residual
          if (EP == 4) v = 0.5f * v * (1.0f + erff(v * 0.70710678118654752f));
          C[(long)mm * N + nn] = v;
        }
      }
    }
  }
}

// ---------------------------------------------------------------------------
// 3) LayerNorm over dim=256: one wave per row, 8 rows per 256-thread block.
// ---------------------------------------------------------------------------
__global__ void layernorm_kernel(const float* __restrict__ x, float* __restrict__ y,
                                 const float* __restrict__ w, const float* __restrict__ b,
                                 int M) {
  int row = blockIdx.x * 8 + (threadIdx.x >> 5);
  int lane = threadIdx.x & 31;
  if (row >= M) return;
  const float* xr = x + (long)row * DIM;
  float vals[8];
  float s = 0.f, s2 = 0.f;
#pragma unroll
  for (int i = 0; i < 8; ++i) {
    float v = xr[lane * 8 + i];
    vals[i] = v; s += v; s2 += v * v;
  }
#pragma unroll
  for (int m = 16; m >= 1; m >>= 1) {
    s  += __shfl_xor(s,  m, 32);
    s2 += __shfl_xor(s2, m, 32);
  }
  float mean = s * (1.f / 256.f);
  float var  = s2 * (1.f / 256.f) - mean * mean;
  float inv  = rsqrtf(var + 1e-5f);
  float* yr = y + (long)row * DIM;
#pragma unroll
  for (int i = 0; i < 8; ++i) {
    int cidx = lane * 8 + i;
    yr[cidx] = (vals[i] - mean) * inv * w[cidx] + b[cidx];
  }
}

// ---------------------------------------------------------------------------
// 4) Fused locality-self-attention (flash-style, strict lower-tri mask).
//    One wave per (b, h, 16-row i-tile). qkv row: [q(0:256) k(256:512)
//    v(512:768)], head h uses cols h*32..h*32+31. Row 0 forced to zero.
// ---------------------------------------------------------------------------
__global__ void lsa_flash_kernel(const float* __restrict__ qkv,
                                 const float* __restrict__ scale_all,
                                 float* __restrict__ out, int layer) {
  __shared__ float P[16 * 18];          // stride 18: 8B-aligned v2f re-reads
  const int bh = blockIdx.y;
  const int b = bh >> 3, h = bh & 7;
  const int i0 = blockIdx.x * 16;
  const int lane = threadIdx.x;
  const int mr = lane & 15, g = lane >> 4;
  const float sc = scale_all[layer * NHEADS + h];
  const float* base = qkv + (long)b * NTOK * 768;

  int qi = i0 + mr; if (qi > NTOK - 1) qi = NTOK - 1;   // clamp (safe: masked)
  const float* qp = base + (long)qi * 768 + h * DH;

  // Q fragments for all 8 K-steps, hoisted out of the j loop.
  v2f qf[8];
#pragma unroll
  for (int kk = 0; kk < 8; ++kk)
    qf[kk] = *(const v2f*)(qp + kk * 4 + 2 * g);

  v8f o0 = {0.f,0.f,0.f,0.f,0.f,0.f,0.f,0.f};
  v8f o1 = o0;
  float mrow[8], lsum[8];
#pragma unroll
  for (int r = 0; r < 8; ++r) { mrow[r] = NEGV; lsum[r] = 0.f; }

  // strict lower-triangular mask: only j-tiles with j0 <= i0+15
  for (int jt = 0; jt <= (int)blockIdx.x; ++jt) {
    const int j0 = jt * 16;
    int kj = j0 + mr; if (kj > NTOK - 1) kj = NTOK - 1;
    const float* kp = base + (long)kj * 768 + 256 + h * DH;

    // preload all 8 K fragments (one load clause), then 8 WMMAs
    v2f kf[8];
#pragma unroll
    for (int kk = 0; kk < 8; ++kk)
      kf[kk] = *(const v2f*)(kp + kk * 4 + 2 * g);

    v8f s = {0.f,0.f,0.f,0.f,0.f,0.f,0.f,0.f};
#pragma unroll
    for (int kk = 0; kk < 8; ++kk)
      s = wmma_f32(qf[kk], kf[kk], s);

    // mask + online softmax (rows live across 16-lane half-groups)
#pragma unroll
    for (int r = 0; r < 8; ++r) {
      int im = i0 + r + 8 * g;
      int jn = j0 + mr;
      float v = s[r] * sc;
      if (jn >= im || im >= NTOK || jn >= NTOK) v = NEGV;
      float tmax = v;
      for (int msk = 1; msk < 16; msk <<= 1)
        tmax = fmaxf(tmax, __shfl_xor(tmax, msk, 16));
      float nm = fmaxf(mrow[r], tmax);
      float alpha = __expf(mrow[r] - nm);   // NEG-NEG==0 -> 1, finite-safe
      float p = __expf(v - nm);
      float ps = p;
      for (int msk = 1; msk < 16; msk <<= 1)
        ps += __shfl_xor(ps, msk, 16);
      lsum[r] = lsum[r] * alpha + ps;
      mrow[r] = nm;
      o0[r] *= alpha;
      o1[r] *= alpha;
      P[(r + 8 * g) * 18 + mr] = p;         // C-layout -> LDS
    }
    __syncthreads();                        // s_wait_dscnt + barrier

    // O += P(16x16) * V(16x32) : preload P (A-layout) + V fragments, then WMMAs
    const float* vb = base + 512 + h * DH;
    v2f pa[4], vf0[4], vf1[4];
#pragma unroll
    for (int kk = 0; kk < 4; ++kk) {
      int k = kk * 4 + 2 * g;
      pa[kk] = *(const v2f*)&P[mr * 18 + k];
      int vr0 = j0 + k;     if (vr0 > NTOK - 1) vr0 = NTOK - 1;  // clamp: p==0
      int vr1 = j0 + k + 1; if (vr1 > NTOK - 1) vr1 = NTOK - 1;
      vf0[kk].x = vb[(long)vr0 * 768 + mr];
      vf0[kk].y = vb[(long)vr1 * 768 + mr];
      vf1[kk].x = vb[(long)vr0 * 768 + 16 + mr];
      vf1[kk].y = vb[(long)vr1 * 768 + 16 + mr];
    }
#pragma unroll
    for (int kk = 0; kk < 4; ++kk) {
      o0 = wmma_f32(pa[kk], vf0[kk], o0);
      o1 = wmma_f32(pa[kk], vf1[kk], o1);
    }
    __syncthreads();
  }

  // normalize + write [B,N,256] at column h*32+d ; row 0 := 0 (ref semantics)
#pragma unroll
  for (int r = 0; r < 8; ++r) {
    int im = i0 + r + 8 * g;
    if (im < NTOK) {
      float inv = (im == 0) ? 0.f : (1.f / lsum[r]);
      long o = ((long)b * NTOK + im) * DIM + h * DH + mr;
      out[o]      = o0[r] * inv;
      out[o + 16] = o1[r] * inv;
    }
  }
}

// ---------------------------------------------------------------------------
// 5) pix[B,N,256] -> out[B,457,457] (un-patch + crop)
// ---------------------------------------------------------------------------
__global__ void scatter_out_kernel(const float* __restrict__ pix,
                                   float* __restrict__ out) {
  int idx = blockIdx.x * 256 + threadIdx.x;
  const int total = BATCH * IMGSZ * IMGSZ;
  if (idx >= total) return;
  int j = idx % IMGSZ;
  int t = idx / IMGSZ;
  int i = t % IMGSZ;
  int b = t / IMGSZ;
  int hp = i >> 4, wp = j >> 4;
  int r = i & 15, cc = j & 15;
  out[idx] = pix[((long)b * NTOK + hp * HP + wp) * 256 + r * 16 + cc];
}

// ---------------------------------------------------------------------------
extern "C" void kernel_launch(void* const* d_in, const int* in_sizes, int n_in,
                              void* d_out, int out_size, void* d_ws, size_t ws_size,
                              hipStream_t stream) {
  const float* x       = (const float*)d_in[0];
  const float* pos_emb = (const float*)d_in[1];
  const float* spt_w   = (const float*)d_in[2];
  const float* spt_b   = (const float*)d_in[3];
  const float* ln1_w   = (const float*)d_in[4];
  const float* ln1_b   = (const float*)d_in[5];
  const float* scale   = (const float*)d_in[6];
  const float* wqkv    = (const float*)d_in[7];
  const float* wout    = (const float*)d_in[8];
  const float* bout    = (const float*)d_in[9];
  const float* ln2_w   = (const float*)d_in[10];
  const float* ln2_b   = (const float*)d_in[11];
  const float* ff1_w   = (const float*)d_in[12];
  const float* ff1_b   = (const float*)d_in[13];
  const float* ff2_w   = (const float*)d_in[14];
  const float* ff2_b   = (const float*)d_in[15];
  const float* pix_w   = (const float*)d_in[16];
  const float* pix_b   = (const float*)d_in[17];
  float* outp = (float*)d_out;

  // workspace layout (floats): T | XN | AO | BIG(U / qkv / h / pix)
  const long TSZ = (long)MTOT * DIM;        // 1,722,368
  float* T   = (float*)d_ws;
  float* XN  = T + TSZ;
  float* AO  = XN + TSZ;
  float* BIG = AO + TSZ;
  float* U    = BIG;                         // MTOT*1280
  float* QKV  = BIG;                         // MTOT*768  (after U is dead)
  float* H    = BIG;                         // MTOT*1024 (after QKV is dead)
  float* PIX  = BIG;                         // MTOT*256  (after H is dead)

  const int gx = (MTOT + 63) / 64;           // 106 row-tiles of 64

  // tokenize: U = gather(x);  T = U @ spt_w^T + spt_b + pos_emb
  build_u_kernel<<<(MTOT * 1280) / 256, 256, 0, stream>>>(x, U);
  gemm_nt_kernel<2><<<dim3(gx, DIM / 128), 256, 0, stream>>>(
      U, spt_w, T, spt_b, pos_emb, MTOT, DIM, 1280);

  for (int d = 0; d < 3; ++d) {
    layernorm_kernel<<<MTOT / 8, 256, 0, stream>>>(
        T, XN, ln1_w + d * DIM, ln1_b + d * DIM, MTOT);
    gemm_nt_kernel<0><<<dim3(gx, 768 / 128), 256, 0, stream>>>(
        XN, wqkv + (long)d * 768 * DIM, QKV, nullptr, nullptr, MTOT, 768, DIM);
    lsa_flash_kernel<<<dim3((NTOK + 15) / 16, BATCH * NHEADS), 32, 0, stream>>>(
        QKV, scale, AO, d);
    gemm_nt_kernel<3><<<dim3(gx, DIM / 128), 256, 0, stream>>>(
        AO, wout + (long)d * DIM * DIM, T, bout + d * DIM, T, MTOT, DIM, DIM);
    layernorm_kernel<<<MTOT / 8, 256, 0, stream>>>(
        T, XN, ln2_w + d * DIM, ln2_b + d * DIM, MTOT);
    gemm_nt_kernel<4><<<dim3(gx, 1024 / 128), 256, 0, stream>>>(
        XN, ff1_w + (long)d * 1024 * DIM, H, ff1_b + d * 1024, nullptr,
        MTOT, 1024, DIM);
    gemm_nt_kernel<3><<<dim3(gx, DIM / 128), 256, 0, stream>>>(
        H, ff2_w + (long)d * DIM * 1024, T, ff2_b + d * DIM, T, MTOT, DIM, 1024);
  }

  gemm_nt_kernel<1><<<dim3(gx, DIM / 128), 256, 0, stream>>>(
      T, pix_w, PIX, pix_b, nullptr, MTOT, DIM, DIM);

  const int outtot = BATCH * IMGSZ * IMGSZ;
  scatter_out_kernel<<<(outtot + 255) / 256, 256, 0, stream>>>(PIX, outp);
}